// LatencyEncoder_44092134260941
// MI455X (gfx1250) — compile-verified
//
#include <hip/hip_runtime.h>
#include <stdint.h>

// Latency encoder: out[row, t] = (sigmoid(x+0.01*noise) > 0.5) ? (t == rne((1-s)*99)) : 0
// 1,048,576 rows of 100 floats -> 419.4 MB streamed out.  Store-BW bound (~18us @ 23.3 TB/s).
// Strategy: stage one 128-row tile (51,200 B) per block in LDS (zero-fill + single
// scalar scatter per row), then drain the tile with CDNA5 async LDS->global b128
// stores (ASYNCcnt path).  The async op applies INST_OFFSET to BOTH the global and
// LDS addresses, so all 25 drain stores share one (vaddr64, ldsaddr32) base pair
// with constant offset:k*2048 -- no per-store address arithmetic.

#define TIME_STEPS      100
#define ROWS_PER_BLOCK  128
#define THREADS         128
#define ROW_BYTES       (TIME_STEPS * 4)                 // 400 B per row
#define TILE_BYTES      (ROWS_PER_BLOCK * ROW_BYTES)     // 51,200 B per block
#define TILE_VEC4       (TILE_BYTES / 16)                // 3,200 float4 per block
#define VEC4_PER_THREAD (TILE_VEC4 / THREADS)            // 25 async stores per thread
#define DRAIN_STRIDE    (THREADS * 16)                   // 2,048 B between stores

__global__ __launch_bounds__(THREADS)
void LatencyEncoder_kernel(const float* __restrict__ x,
                           const float* __restrict__ noise,
                           float* __restrict__ out)
{
    extern __shared__ float4 tile[];                     // TILE_VEC4 float4s, 16B aligned
    const int tid = threadIdx.x;

    // ---- per-row compute (coalesced b32 loads, one row per lane) ----
    const long long row = (long long)blockIdx.x * ROWS_PER_BLOCK + tid;
    const float xv = x[row];
    const float nv = noise[row];
    const float xn = fmaf(nv, 0.01f, xv);
    const float e  = __expf(-xn);                        // v_exp_f32
    const float s  = __builtin_amdgcn_rcpf(1.0f + e);    // v_rcp_f32: sigmoid
    int lat = __float2int_rn((1.0f - s) * 99.0f);        // round-to-nearest-even == jnp.round
    lat = min(max(lat, 0), 99);
    const bool spike = (s > 0.5f);

    // ---- stage row into LDS: zero-fill with b128, then one predicated scatter ----
    float4* myrow = tile + tid * (TIME_STEPS / 4);       // 25 float4 per row
    const float4 z4 = make_float4(0.0f, 0.0f, 0.0f, 0.0f);
#pragma unroll
    for (int j = 0; j < TIME_STEPS / 4; ++j)
        myrow[j] = z4;                                   // ds_store_b128 x25, folded offsets
    if (spike)
        ((float*)myrow)[lat] = 1.0f;                     // ds_store_b32 (in-order after b128s)

    __syncthreads();                                     // DS writes visible block-wide

    // ---- drain tile with CDNA5 async LDS->global stores (coalesced b128) ----
    // One base address pair per thread; per-iteration displacement goes in the
    // instruction's 24-bit IOFFSET (applied to both LDS and global sides).
    const uint32_t la = (uint32_t)(uintptr_t)tile + (uint32_t)tid * 16u;
    const uint64_t ga = (uint64_t)(uintptr_t)out
                      + (uint64_t)blockIdx.x * TILE_BYTES
                      + (uint32_t)tid * 16u;
#pragma unroll
    for (int k = 0; k < VEC4_PER_THREAD; ++k) {
        asm volatile("global_store_async_from_lds_b128 %0, %1, off offset:%2"
                     :: "v"(ga), "v"(la), "i"(k * DRAIN_STRIDE)
                     : "memory");
    }
    // Retire all async transfers before the wave ends (ASYNCcnt DMA path).
    asm volatile("s_wait_asynccnt 0x0" ::: "memory");
}

extern "C" void kernel_launch(void* const* d_in, const int* in_sizes, int n_in,
                              void* d_out, int out_size, void* d_ws, size_t ws_size,
                              hipStream_t stream) {
    (void)in_sizes; (void)n_in; (void)d_ws; (void)ws_size; (void)out_size;
    const float* x     = (const float*)d_in[0];
    const float* noise = (const float*)d_in[1];
    float* out         = (float*)d_out;

    // 1,048,576 rows / 128 rows per block = 8192 blocks; 51,200 B dynamic LDS each.
    dim3 grid(8192), block(THREADS);
    LatencyEncoder_kernel<<<grid, block, TILE_BYTES, stream>>>(x, noise, out);
}